// NSEInt_45509473469110
// MI455X (gfx1250) — compile-verified
//
#include <hip/hip_runtime.h>
#include <math.h>
#include <stdint.h>

#define NVOX  (64*64*64)
#define NSTEP 7
#define SEPS  1e-4f

// ---------------- CDNA5 TDM availability ----------------
#if defined(__gfx1250__) && __has_builtin(__builtin_amdgcn_tensor_load_to_lds) && \
    __has_builtin(__builtin_amdgcn_s_wait_tensorcnt)
#define HAVE_TDM 1
#else
#define HAVE_TDM 0
#endif

typedef unsigned int u32x4 __attribute__((ext_vector_type(4)));
typedef int          i32x4 __attribute__((ext_vector_type(4)));
typedef int          i32x8 __attribute__((ext_vector_type(8)));

// ---------- SE(3) helpers (faithful to the reference's branch structure) ----------

__device__ __forceinline__ void se3_exp6(const float xi[6], float R[9], float t[3]) {
    const float v0 = xi[0], v1 = xi[1], v2 = xi[2];
    const float w0 = xi[3], w1 = xi[4], w2 = xi[5];
    const float t2 = w0*w0 + w1*w1 + w2*w2;
    const float theta = sqrtf(t2);
    const bool  sm = theta < SEPS;
    const float ts = sm ? 1.0f : theta;
    const float s  = sinf(ts);
    const float c  = cosf(ts);
    const float A  = sm ? 1.0f - t2*(1.0f/6.0f)          : s/ts;
    const float B  = sm ? 0.5f - t2*(1.0f/24.0f)         : (1.0f - c)/(ts*ts);
    const float C  = sm ? (1.0f/6.0f) - t2*(1.0f/120.0f) : (ts - s)/(ts*ts*ts);
    const float W2_00 = -(w1*w1 + w2*w2);
    const float W2_11 = -(w0*w0 + w2*w2);
    const float W2_22 = -(w0*w0 + w1*w1);
    const float W2_01 = w0*w1, W2_02 = w0*w2, W2_12 = w1*w2;
    R[0] = 1.0f + B*W2_00;  R[1] = -A*w2 + B*W2_01;  R[2] =  A*w1 + B*W2_02;
    R[3] =  A*w2 + B*W2_01; R[4] = 1.0f + B*W2_11;   R[5] = -A*w0 + B*W2_12;
    R[6] = -A*w1 + B*W2_02; R[7] =  A*w0 + B*W2_12;  R[8] = 1.0f + B*W2_22;
    const float V00 = 1.0f + C*W2_00,  V01 = -B*w2 + C*W2_01, V02 =  B*w1 + C*W2_02;
    const float V10 =  B*w2 + C*W2_01, V11 = 1.0f + C*W2_11,  V12 = -B*w0 + C*W2_12;
    const float V20 = -B*w1 + C*W2_02, V21 =  B*w0 + C*W2_12, V22 = 1.0f + C*W2_22;
    t[0] = V00*v0 + V01*v1 + V02*v2;
    t[1] = V10*v0 + V11*v1 + V12*v2;
    t[2] = V20*v0 + V21*v1 + V22*v2;
}

__device__ __forceinline__ void se3_log6(const float R[9], const float t[3], float xi[6]) {
    const float tr   = R[0] + R[4] + R[8];
    float cosv = (tr - 1.0f)*0.5f;
    cosv = fminf(fmaxf(cosv, -1.0f + 1e-6f), 1.0f - 1e-6f);
    const float theta = acosf(cosv);
    const bool  sm = theta < SEPS;
    const float ts = sm ? 1.0f : theta;
    const float t2 = theta*theta;
    const float s  = sinf(ts);
    const float c  = cosf(ts);
    const float fac = sm ? 1.0f + t2*(1.0f/6.0f) : ts/s;
    const float w0 = 0.5f*(R[7] - R[5])*fac;
    const float w1 = 0.5f*(R[2] - R[6])*fac;
    const float w2 = 0.5f*(R[3] - R[1])*fac;
    const float A  = sm ? 1.0f - t2*(1.0f/6.0f)  : s/ts;
    const float B  = sm ? 0.5f - t2*(1.0f/24.0f) : (1.0f - c)/(ts*ts);
    const float coef = sm ? (1.0f/12.0f) : (1.0f - A/(2.0f*B))/t2;
    const float W2_00 = -(w1*w1 + w2*w2);
    const float W2_11 = -(w0*w0 + w2*w2);
    const float W2_22 = -(w0*w0 + w1*w1);
    const float W2_01 = w0*w1, W2_02 = w0*w2, W2_12 = w1*w2;
    const float Vi00 = 1.0f + coef*W2_00,      Vi01 =  0.5f*w2 + coef*W2_01, Vi02 = -0.5f*w1 + coef*W2_02;
    const float Vi10 = -0.5f*w2 + coef*W2_01,  Vi11 = 1.0f + coef*W2_11,     Vi12 =  0.5f*w0 + coef*W2_12;
    const float Vi20 =  0.5f*w1 + coef*W2_02,  Vi21 = -0.5f*w0 + coef*W2_12, Vi22 = 1.0f + coef*W2_22;
    xi[0] = Vi00*t[0] + Vi01*t[1] + Vi02*t[2];
    xi[1] = Vi10*t[0] + Vi11*t[1] + Vi12*t[2];
    xi[2] = Vi20*t[0] + Vi21*t[1] + Vi22*t[2];
    xi[3] = w0; xi[4] = w1; xi[5] = w2;
}

// ---------- init: vec_f = vec * 2^-7 (voxel-major, 32B padded), mulMat = se3_exp ----------
// TDM: one lane DMA-stages this block's six strided channel slabs of `vin`
// (2-D tile: 256 floats x 6 rows, row stride NVOX) into LDS. The compute path
// keeps the authoritative direct loads; the DMA is an L2-warming prefetch of
// exactly the lines this block reads, and exercises the gfx1250 TDM path.

__global__ __launch_bounds__(256)
void se3_init_kernel(const float* __restrict__ vin,
                     float* __restrict__ vecf,
                     float* __restrict__ mats) {
#if HAVE_TDM
    __shared__ float tdm_stage[6*256];
    if (threadIdx.x == 0) {
        const uint64_t ga  = (uint64_t)(uintptr_t)(vin + (size_t)blockIdx.x*256u);
        const unsigned lds = (unsigned)(unsigned long long)
                             (__attribute__((address_space(3))) void*)tdm_stage;
        // D# group0: count=1 | lds_addr | global_addr[56:0] | type=2 at [127:126]
        u32x4 g0 = { 1u, lds, (unsigned)ga,
                     (unsigned)((ga >> 32) & 0x01FFFFFFu) | 0x80000000u };
        // D# group1: data_size=4B; tensor_dim0=256; tensor_dim1=6; tile_dim0=256;
        //            tile_dim1=6; tensor_dim0_stride=NVOX (channel plane stride)
        i32x8 g1 = { (int)(2u << 16),        // data_size=2 (4B) at [17:16]
                     (int)(256u << 16),      // tensor_dim0[15:0] at [63:48]
                     (int)(6u << 16),        // tensor_dim0 hi=0 | tensor_dim1[15:0] at [111:96... (95:80)]
                     (int)(256u << 16),      // tensor_dim1 hi=0 | tile_dim0 at [127:112]
                     6,                      // tile_dim1=6 | tile_dim2=0
                     (int)NVOX,              // tensor_dim0_stride low 32
                     0, 0 };
        i32x4 g2 = {0,0,0,0}, g3 = {0,0,0,0};
#if __clang_major__ >= 23
        i32x8 g4 = {0,0,0,0,0,0,0,0};
        __builtin_amdgcn_tensor_load_to_lds(g0, g1, g2, g3, g4, 0);
#else
        __builtin_amdgcn_tensor_load_to_lds(g0, g1, g2, g3, 0);
#endif
        __builtin_amdgcn_s_wait_tensorcnt(0);
    }
#endif

    const int n = blockIdx.x*blockDim.x + threadIdx.x;
    if (n >= NVOX) return;
    const float scale = 1.0f/128.0f;
    float xi[6];
#pragma unroll
    for (int c = 0; c < 6; ++c) xi[c] = vin[c*NVOX + n]*scale;

    float4* v4 = (float4*)(vecf + (size_t)n*8);
    v4[0] = make_float4(xi[0], xi[1], xi[2], xi[3]);
    ((float2*)v4)[2] = make_float2(xi[4], xi[5]);   // floats 4..5; 6..7 = pad (never read)

    float R[9], t[3];
    se3_exp6(xi, R, t);
    float4* m = (float4*)(mats + (size_t)n*12);
    m[0] = make_float4(R[0], R[1], R[2], t[0]);
    m[1] = make_float4(R[3], R[4], R[5], t[1]);
    m[2] = make_float4(R[6], R[7], R[8], t[2]);
}

// ---------- one squaring step (FINAL: write packed (N,6) rows to d_out) ----------

template<bool FINAL>
__global__ __launch_bounds__(256)
void se3_step_kernel(const float* __restrict__ vecf_in,
                     float* __restrict__ vecf_out,
                     float* __restrict__ mats) {
    const int n = blockIdx.x*blockDim.x + threadIdx.x;
    if (n >= NVOX) return;

    float4* mrow = (float4*)(mats + (size_t)n*12);
    __builtin_prefetch(mrow, 0, 3);              // global_prefetch_b8 (gfx1250)
    const float4 m0 = mrow[0], m1 = mrow[1], m2 = mrow[2];

    const int wx = n & 63, hy = (n >> 6) & 63, dz = n >> 12;
    const float p0 = (float)(2*dz - 63) / 63.0f;   // d-axis
    const float p1 = (float)(2*hy - 63) / 63.0f;   // h-axis
    const float p2 = (float)(2*wx - 63) / 63.0f;   // w-axis

    const float nl0 = m0.x*p0 + m0.y*p1 + m0.z*p2 + m0.w;
    const float nl1 = m1.x*p0 + m1.y*p1 + m1.z*p2 + m1.w;
    const float nl2 = m2.x*p0 + m2.y*p1 + m2.z*p2 + m2.w;

    // samp = reversed -> (x <- nl2, y <- nl1, z <- nl0); denom == 1 for 64^3
    const float gx = fminf(fmaxf((nl2 + 1.0f)*0.5f*63.0f, 0.0f), 63.0f);
    const float gy = fminf(fmaxf((nl1 + 1.0f)*0.5f*63.0f, 0.0f), 63.0f);
    const float gz = fminf(fmaxf((nl0 + 1.0f)*0.5f*63.0f, 0.0f), 63.0f);
    const float x0f = floorf(gx), y0f = floorf(gy), z0f = floorf(gz);
    const float fx = gx - x0f, fy = gy - y0f, fz = gz - z0f;
    const int x0 = (int)x0f, y0 = (int)y0f, z0 = (int)z0f;
    const int x1 = min(x0 + 1, 63), y1 = min(y0 + 1, 63), z1 = min(z0 + 1, 63);
    const float ox = 1.0f - fx, oy = 1.0f - fy, oz = 1.0f - fz;

    float a0 = 0.f, a1 = 0.f, a2 = 0.f, a3 = 0.f, a4 = 0.f, a5 = 0.f;
#pragma unroll
    for (int cz = 0; cz < 2; ++cz) {
        const int   zi = cz ? z1 : z0;
        const float wz = cz ? fz : oz;
#pragma unroll
        for (int cy = 0; cy < 2; ++cy) {
            const int   yi  = cy ? y1 : y0;
            const float wyz = (cy ? fy : oy)*wz;
#pragma unroll
            for (int cx = 0; cx < 2; ++cx) {
                const int   xi = cx ? x1 : x0;
                const float w  = (cx ? fx : ox)*wyz;
                // padded 32B rows: 16B-aligned B128 + B64 per corner
                const float* base = vecf_in + (size_t)(((zi*64 + yi)*64 + xi))*8;
                const float4 va = *(const float4*)base;
                const float2 vb = *(const float2*)(base + 4);
                a0 += w*va.x; a1 += w*va.y;
                a2 += w*va.z; a3 += w*va.w;
                a4 += w*vb.x; a5 += w*vb.y;
            }
        }
    }

    const float interp[6] = {a0, a1, a2, a3, a4, a5};
    float R2[9], t2v[3];
    se3_exp6(interp, R2, t2v);

    // mulMat = Mat2 @ mulMat  (4x4 with implicit bottom row)
    const float r1[9] = {m0.x, m0.y, m0.z, m1.x, m1.y, m1.z, m2.x, m2.y, m2.z};
    const float t1[3] = {m0.w, m1.w, m2.w};
    float Rn[9], tn[3];
#pragma unroll
    for (int i = 0; i < 3; ++i) {
        const float ri0 = R2[i*3+0], ri1 = R2[i*3+1], ri2 = R2[i*3+2];
#pragma unroll
        for (int j = 0; j < 3; ++j)
            Rn[i*3+j] = ri0*r1[0*3+j] + ri1*r1[1*3+j] + ri2*r1[2*3+j];
        tn[i] = ri0*t1[0] + ri1*t1[1] + ri2*t1[2] + t2v[i];
    }

    mrow[0] = make_float4(Rn[0], Rn[1], Rn[2], tn[0]);
    mrow[1] = make_float4(Rn[3], Rn[4], Rn[5], tn[1]);
    mrow[2] = make_float4(Rn[6], Rn[7], Rn[8], tn[2]);

    float xo[6];
    se3_log6(Rn, tn, xo);
    if (FINAL) {
        float2* vo = (float2*)(vecf_out + (size_t)n*6);   // packed (N,6) output
        vo[0] = make_float2(xo[0], xo[1]);
        vo[1] = make_float2(xo[2], xo[3]);
        vo[2] = make_float2(xo[4], xo[5]);
    } else {
        float4* vo = (float4*)(vecf_out + (size_t)n*8);   // padded intermediate
        vo[0] = make_float4(xo[0], xo[1], xo[2], xo[3]);
        ((float2*)vo)[2] = make_float2(xo[4], xo[5]);
    }
}

// ---------- launcher ----------

extern "C" void kernel_launch(void* const* d_in, const int* in_sizes, int n_in,
                              void* d_out, int out_size, void* d_ws, size_t ws_size,
                              hipStream_t stream) {
    (void)in_sizes; (void)n_in; (void)out_size; (void)ws_size;
    const float* vin = (const float*)d_in[0];
    float* out  = (float*)d_out;
    float* ws   = (float*)d_ws;

    float* vecfA = ws;                           // NVOX*8 floats (padded rows)
    float* vecfB = ws + (size_t)NVOX*8;          // NVOX*8 floats
    float* mats  = ws + (size_t)NVOX*16;         // NVOX*12 floats

    const dim3 blk(256), grd((NVOX + 255)/256);
    se3_init_kernel<<<grd, blk, 0, stream>>>(vin, vecfA, mats);

    float* bufs[2] = {vecfA, vecfB};
    for (int s = 0; s < NSTEP; ++s) {
        const float* src = bufs[s & 1];
        if (s == NSTEP - 1)
            se3_step_kernel<true ><<<grd, blk, 0, stream>>>(src, out, mats);
        else
            se3_step_kernel<false><<<grd, blk, 0, stream>>>(src, bufs[(s + 1) & 1], mats);
    }
}